// ReprojectionLayer_70909910057050
// MI455X (gfx1250) — compile-verified
//
#include <hip/hip_runtime.h>
#include <hip/hip_bf16.h>

#define C_CAM 12
#define J_NUM 23
#define HS 130
#define G2 40
#define G 80
#define G2_3 (G2*G2*G2)      // 64000
#define G_3  (G*G*G)         // 512000

typedef float v2f __attribute__((ext_vector_type(2)));
typedef float v8f __attribute__((ext_vector_type(8)));

// ---------------------------------------------------------------------------
// Kernel A: project the 40^3 grid through all 12 cameras with WMMA
// (D = A(16x4 points) x B(4x16 packed camera cols)), then distort, clip,
// and store per-camera v1/v2 coordinate fields to workspace.
// One wave (32 threads) per 16 grid points -> 4000 blocks.
// ---------------------------------------------------------------------------
__global__ __launch_bounds__(32)
void ReprojectionProjectKernel(const float* __restrict__ center,
                               const float* __restrict__ cHM,
                               const float* __restrict__ P,
                               const float* __restrict__ Km,
                               const float* __restrict__ Dm,
                               float* __restrict__ wsv1,
                               float* __restrict__ wsv2) {
    __shared__ float lds_proj[16 * 36];   // [point_local][cam*3+e]

    const int lane = threadIdx.x;         // 0..31, wave32
    const int n    = lane & 15;
    const int half = lane >> 4;
    const int p0   = blockIdx.x * 16;

    // A fragment: 16 points x 4 homogeneous coords.
    // ISA layout (16x4 f32 A): lanes 0-15 M=lane, VGPR0=K0,VGPR1=K1;
    //                          lanes 16-31 M=lane-16, VGPR0=K2,VGPR1=K3.
    const int p = p0 + n;
    const int gi = p / 1600;
    const int gj = (p / 40) % 40;
    const int gk = p % 40;
    float comp[4];
    comp[0] = (float)(gi - 20) * 4.0f + center[0];
    comp[1] = (float)(gj - 20) * 4.0f + center[1];
    comp[2] = (float)(gk - 20) * 4.0f + center[2];
    comp[3] = 1.0f;
    v2f a;
    a.x = comp[2 * half];
    a.y = comp[2 * half + 1];

    // Three WMMAs: camera groups {0..4}, {5..9}, {10..11}, 3 cols per camera.
    // B layout (4x16 f32 B, mirrors A): lanes 0-15 K=0(V0)/1(V1), lanes 16-31 K=2/3.
    const int caml = n / 3;
    const int e    = n - caml * 3;
    for (int g = 0; g < 3; ++g) {
        const int cg    = g * 5;
        const int ncams = (g == 2) ? 2 : 5;
        const int ncols = ncams * 3;
        const bool valid = (n < ncols);
        const int cam = min(cg + caml, C_CAM - 1);     // clamp: keep loads in-bounds
        v2f b;
        b.x = valid ? P[cam * 12 + (2 * half) * 3 + e]     : 0.0f;
        b.y = valid ? P[cam * 12 + (2 * half + 1) * 3 + e] : 0.0f;
        v8f cz = {};
        // D[m][n] = sum_k A[m][k]*B[k][n]  ->  proj[cam(n)][point m][e(n)]
        v8f d = __builtin_amdgcn_wmma_f32_16x16x4_f32(
            /*neg_a=*/false, a, /*neg_b=*/false, b,
            /*c_mod=*/(short)0, cz, /*reuse_a=*/false, /*reuse_b=*/false);
        if (valid) {
            // D layout: VGPR r holds M=r (lanes<16) / M=r+8 (lanes>=16), N=lane&15
#pragma unroll
            for (int r = 0; r < 8; ++r)
                lds_proj[(8 * half + r) * 36 + cam * 3 + e] = d[r];
        }
    }
    __syncthreads();

    // Phase 2: 16 points x 12 cams = 192 (point,cam) pairs, 6 per lane.
    for (int t = 0; t < 6; ++t) {
        const int pair = lane + 32 * t;
        const int pl   = pair / C_CAM;
        const int cam  = pair % C_CAM;
        const float px = lds_proj[pl * 36 + cam * 3 + 0];
        const float py = lds_proj[pl * 36 + cam * 3 + 1];
        const float pz = lds_proj[pl * 36 + cam * 3 + 2];

        const float fx = Km[cam * 9 + 0], fy = Km[cam * 9 + 4];
        const float cx = Km[cam * 9 + 6], cy = Km[cam * 9 + 7];
        const float k1 = Dm[cam * 5 + 0], k2 = Dm[cam * 5 + 1];
        const float chx = cHM[cam * 2 + 0], chy = cHM[cam * 2 + 1];

        float v1 = px / pz - cx;
        float v2 = py / pz - cy;
        const float t1 = v1 / fx, t2 = v2 / fy;
        const float r2 = t1 * t1 + t2 * t2;
        const float dist = 1.0f + (k1 + k2 * r2) * r2;
        v1 = v1 * dist + cx;
        v2 = v2 * dist + cy;
        v1 = fminf(fmaxf(v1, chx - (float)(HS - 1)), chx + (float)(HS - 2)) - chx + (float)(HS - 1);
        v2 = fminf(fmaxf(v2, chy - (float)(HS - 1)), chy + (float)(HS - 2)) - chy + (float)(HS - 1);

        const int pp = p0 + pl;
        wsv1[cam * G2_3 + pp] = v1;
        wsv2[cam * G2_3 + pp] = v2;
    }
}

// ---------------------------------------------------------------------------
// Kernel B: fused trilinear coordinate upsample (40^3 -> 80^3) + heatmap
// gather + camera mean. Blocks = 8^3 voxel tiles; per camera the 6^3 coarse
// coordinate support region is staged in LDS. 23 joint accumulators in VGPRs.
// ---------------------------------------------------------------------------
__global__ __launch_bounds__(512)
void ReprojectionGatherKernel(const float* __restrict__ heat,
                              const float* __restrict__ wsv1,
                              const float* __restrict__ wsv2,
                              float* __restrict__ out) {
    __shared__ float l1[216];
    __shared__ float l2[216];

    const int tid = threadIdx.x;
    const int tz = tid & 7, ty = (tid >> 3) & 7, tx = tid >> 6;
    const int bi = blockIdx.x, bj = blockIdx.y, bk = blockIdx.z;
    const int vi = bi * 8 + tx, vj = bj * 8 + ty, vk = bk * 8 + tz;

    const int cbi = 4 * bi - 1, cbj = 4 * bj - 1, cbk = 4 * bk - 1;

    // torch/jax align_corners=false taps: pos = clamp(p/2 - 0.25, 0, 39)
    auto taps = [](int p, int cb, int& r0, int& r1, float& w) {
        float pos = fminf(fmaxf((float)p * 0.5f - 0.25f, 0.0f), 39.0f);
        int i0 = (int)pos;           // pos >= 0: trunc == floor
        w = pos - (float)i0;
        int i1 = min(i0 + 1, 39);
        r0 = i0 - cb;                // in [0,5] relative to LDS tile
        r1 = i1 - cb;
    };
    int ri0, ri1, rj0, rj1, rk0, rk1;
    float wi, wj, wk;
    taps(vi, cbi, ri0, ri1, wi);
    taps(vj, cbj, rj0, rj1, wj);
    taps(vk, cbk, rk0, rk1, wk);

    float acc[J_NUM];
#pragma unroll
    for (int j = 0; j < J_NUM; ++j) acc[j] = 0.0f;

    for (int cam = 0; cam < C_CAM; ++cam) {
        __syncthreads();
        if (tid < 216) {
            const int oi = tid / 36, oj = (tid / 6) % 6, ok = tid % 6;
            const int gi = min(max(cbi + oi, 0), 39);
            const int gj = min(max(cbj + oj, 0), 39);
            const int gk = min(max(cbk + ok, 0), 39);
            const int off = cam * G2_3 + gi * 1600 + gj * 40 + gk;
            l1[tid] = wsv1[off];
            l2[tid] = wsv2[off];
        }
        __syncthreads();

        // lerp axis-1 (i) first, then axis-2 (j), then axis-3 (k) — matches
        // the reference's separable upsample order.
        const int b00 = ri0 * 36, b01 = ri1 * 36;
        float ai[2][2], bi2[2][2];
#pragma unroll
        for (int jb = 0; jb < 2; ++jb) {
            const int rj = (jb ? rj1 : rj0) * 6;
#pragma unroll
            for (int kc = 0; kc < 2; ++kc) {
                const int rk = kc ? rk1 : rk0;
                const int o0 = b00 + rj + rk, o1 = b01 + rj + rk;
                ai[jb][kc]  = l1[o0] * (1.0f - wi) + l1[o1] * wi;
                bi2[jb][kc] = l2[o0] * (1.0f - wi) + l2[o1] * wi;
            }
        }
        const float a0 = ai[0][0] * (1.0f - wj) + ai[1][0] * wj;
        const float a1 = ai[0][1] * (1.0f - wj) + ai[1][1] * wj;
        const float v1u = a0 * (1.0f - wk) + a1 * wk;
        const float c0 = bi2[0][0] * (1.0f - wj) + bi2[1][0] * wj;
        const float c1 = bi2[0][1] * (1.0f - wj) + bi2[1][1] * wj;
        const float v2u = c0 * (1.0f - wk) + c1 * wk;

        const int ix = (int)(v1u * 0.5f);   // [0,128]
        const int iy = (int)(v2u * 0.5f);   // [0,128]
        const int res = iy * HS + ix;

        // heatmaps layout (C, J, HS, HS): all 23 joints share one index.
        const float* hp = heat + cam * (J_NUM * HS * HS) + res;
#pragma unroll
        for (int j = 0; j < J_NUM; ++j)
            acc[j] += hp[j * (HS * HS)];
    }

    const int vox = vi * (G * G) + vj * G + vk;
    const float inv = 1.0f / (float)C_CAM;
#pragma unroll
    for (int j = 0; j < J_NUM; ++j)
        out[j * G_3 + vox] = acc[j] * inv;
}

// ---------------------------------------------------------------------------
extern "C" void kernel_launch(void* const* d_in, const int* in_sizes, int n_in,
                              void* d_out, int out_size, void* d_ws, size_t ws_size,
                              hipStream_t stream) {
    const float* heat   = (const float*)d_in[0];   // (1,C,J,HS,HS)
    const float* center = (const float*)d_in[1];   // (1,3)
    const float* cHM    = (const float*)d_in[2];   // (1,C,2)
    const float* P      = (const float*)d_in[3];   // (1,C,4,3)
    const float* Km     = (const float*)d_in[4];   // (1,C,3,3)
    const float* Dm     = (const float*)d_in[5];   // (1,C,1,5)
    float* out = (float*)d_out;                    // (1,J,80,80,80)

    float* wsv1 = (float*)d_ws;                    // C * 40^3 floats
    float* wsv2 = wsv1 + C_CAM * G2_3;             // C * 40^3 floats (6.1 MB total)

    ReprojectionProjectKernel<<<G2_3 / 16, 32, 0, stream>>>(
        center, cHM, P, Km, Dm, wsv1, wsv2);

    dim3 grid(G / 8, G / 8, G / 8);                // 10x10x10 tiles
    ReprojectionGatherKernel<<<grid, 512, 0, stream>>>(heat, wsv1, wsv2, out);
}